// GCN_58686433132930
// MI455X (gfx1250) — compile-verified
//
#include <hip/hip_runtime.h>

// ---------------------------------------------------------------------------
// GCN forward for MI455X (gfx1250, wave32).
//   k  = relu(seq@Wfc^T + mm*seq@Wfc2^T)        -- V_WMMA_F32_16X16X4_F32
//   3x: kappa = colsum(k); agg = A@k (COO, L2-resident atomics);
//       kn = agg + 0.5k; D = rowsum(kn/kappa);
//       k  = kn/(D*kappa) + 0.1kn + 0.2k0
//   out = prelu(k + bias)
// A tile staged to LDS via GLOBAL_LOAD_ASYNC_TO_LDS (ASYNCcnt) when available.
// ---------------------------------------------------------------------------

typedef __attribute__((ext_vector_type(2))) float v2f;
typedef __attribute__((ext_vector_type(8))) float v8f;
typedef __attribute__((ext_vector_type(4))) int   v4i;

#define IN_FT  256
#define OUT_FT 128
#define A_PAD  260   // 260 % 64 == 4 -> 16 row-parallel lanes hit distinct banks

#if defined(__HIP_DEVICE_COMPILE__) &&                                        \
    __has_builtin(__builtin_amdgcn_global_load_async_to_lds_b128) &&          \
    __has_builtin(__builtin_amdgcn_s_wait_asynccnt)
#define USE_ASYNC_LDS 1
typedef __attribute__((address_space(1))) v4i as1_v4i;   // global (__device__)
typedef __attribute__((address_space(3))) v4i as3_v4i;   // LDS    (__shared__)
#else
#define USE_ASYNC_LDS 0
#endif

// ---------------------------------------------------------------------------
// Front GEMM: one block = 16 output rows, 8 waves = 8 column tiles of 16.
// f32 WMMA fragment layout (ISA 7.12.2):
//   A 16x4 : lane l<16 -> row M=l, (K=k0,k0+1); lane l>=16 -> row M=l-16,(k0+2,k0+3)
//   B 4x16 : lane l<16 -> col N=l, (K=k0,k0+1); lane l>=16 -> col N=l-16,(k0+2,k0+3)
//   C/D    : vgpr j -> M = j + 8*(lane>=16), N = lane&15
// B[k,n] = W[n,k] row-major => each lane's B fragment is 2 contiguous floats.
// ---------------------------------------------------------------------------
__global__ __launch_bounds__(256) void gcn_front_gemm(
    const float* __restrict__ seq, const float* __restrict__ Wfc,
    const float* __restrict__ Wfc2, const float* __restrict__ mm,
    float* __restrict__ k, float* __restrict__ k0, float* __restrict__ kappa0,
    int N)
{
  __shared__ float atile[16][A_PAD];           // 16 x 256 A tile, bank-padded

  const int wave = threadIdx.x >> 5;
  const int lane = threadIdx.x & 31;
  const int m    = lane & 15;
  const int hi   = lane >> 4;                  // half-wave select
  const int rowBase = blockIdx.x * 16;

  // ---- stage A tile (16 rows x 256 f32 = 16 KB) into LDS -----------------
  // 1024 b128 chunks, 4 per thread.
#pragma unroll
  for (int i = 0; i < 4; ++i) {
    int ch = threadIdx.x + 256 * i;            // chunk id
    int r  = ch >> 6;                          // tile row 0..15
    int c4 = (ch & 63) * 4;                    // float column 0..252
    int gr = rowBase + r;
    if (gr >= N) gr = N - 1;                   // clamp (tail tiles)
    const float* src = seq + (size_t)gr * IN_FT + c4;
    float*       dst = &atile[r][c4];
#if USE_ASYNC_LDS
    __builtin_amdgcn_global_load_async_to_lds_b128(
        (as1_v4i*)src, (as3_v4i*)dst, 0, 0);
#else
    *(float4*)dst = *(const float4*)src;       // fallback: load + ds_store
#endif
  }
#if USE_ASYNC_LDS
  __builtin_amdgcn_s_wait_asynccnt(0);
#endif
  __syncthreads();

  const int    col   = wave * 16 + m;          // global output column
  const float* b1Ptr = Wfc  + (size_t)col * IN_FT + hi * 2;
  const float* b2Ptr = Wfc2 + (size_t)col * IN_FT + hi * 2;
  const float* aLds  = &atile[m][hi * 2];

  v8f acc1 = {}; v8f acc2 = {};
#pragma unroll 4
  for (int kk = 0; kk < IN_FT; kk += 4) {
    v2f a  = *(const v2f*)(aLds  + kk);        // ds_load_b64, conflict-free
    v2f b1 = *(const v2f*)(b1Ptr + kk);
    v2f b2 = *(const v2f*)(b2Ptr + kk);
    acc1 = __builtin_amdgcn_wmma_f32_16x16x4_f32(false, a, false, b1,
                                                 (short)0, acc1, false, false);
    acc2 = __builtin_amdgcn_wmma_f32_16x16x4_f32(false, a, false, b2,
                                                 (short)0, acc2, false, false);
  }

  const float mmv = mm[0];
  float colsum = 0.f;
  if (rowBase + 16 <= N) {                     // fast path: full tile
#pragma unroll
    for (int j = 0; j < 8; ++j) {
      float r = fmaxf(fmaf(mmv, acc2[j], acc1[j]), 0.f);
      size_t idx = (size_t)(rowBase + hi * 8 + j) * OUT_FT + col;
      k[idx]  = r;
      k0[idx] = r;
      colsum += r;
    }
  } else {                                     // tail tile: guarded stores
#pragma unroll
    for (int j = 0; j < 8; ++j) {
      float r = fmaxf(fmaf(mmv, acc2[j], acc1[j]), 0.f);
      int row = rowBase + hi * 8 + j;
      if (row < N) {
        size_t idx = (size_t)row * OUT_FT + col;
        k[idx]  = r;
        k0[idx] = r;
        colsum += r;
      }
    }
  }
  // column partial for kappa_0: combine half-waves, 16 atomics per wave
  colsum += __shfl_xor(colsum, 16, 32);
  if (hi == 0) atomicAdd(&kappa0[col], colsum);
}

// ---------------------------------------------------------------------------
// COO scatter: agg[row] += k[col] * w. One wave covers 32 consecutive
// features of one edge (uniform row/col/w per wave; coalesced L2 atomics).
// k[N,128] = 51 MB fits in the 192 MB L2, so gathers stay on-chip.
// ---------------------------------------------------------------------------
__global__ __launch_bounds__(256) void gcn_scatter(
    const float* __restrict__ k, const int* __restrict__ erow,
    const int* __restrict__ ecol, const float* __restrict__ ew,
    float* __restrict__ agg, int E)
{
  long long gid = (long long)blockIdx.x * 256 + threadIdx.x;
  int e = (int)(gid >> 7);
  int f = (int)(gid & (OUT_FT - 1));
  if (e >= E) return;
  int r = erow[e];
  int c = ecol[e];
  float w = ew[e];
  atomicAdd(&agg[(size_t)r * OUT_FT + f], k[(size_t)c * OUT_FT + f] * w);
}

// ---------------------------------------------------------------------------
// Per-step update. One wave per row (4 features/lane), 8 waves x 16 rows per
// block. Row-sum D via wave32 shfl_xor reduction. Column sums for the next
// kappa accumulate in registers -> LDS -> 128 global atomics per block.
// Last step fuses +bias and PReLU and writes the final output.
// ---------------------------------------------------------------------------
__global__ __launch_bounds__(256) void gcn_update(
    float* __restrict__ k, const float* __restrict__ agg,
    const float* __restrict__ k0, const float* __restrict__ kap_cur,
    float* __restrict__ kap_next, const float* __restrict__ bias,
    const float* __restrict__ alpha, int N, int is_last)
{
  __shared__ float cs[OUT_FT];
  const int wave = threadIdx.x >> 5;
  const int lane = threadIdx.x & 31;
  const int f0   = lane * 4;

  if (threadIdx.x < OUT_FT) cs[threadIdx.x] = 0.f;
  __syncthreads();

  const float4 kap  = *(const float4*)(kap_cur + f0);
  const float4 ikap = make_float4(1.f / kap.x, 1.f / kap.y,
                                  1.f / kap.z, 1.f / kap.w);
  float4 bs = make_float4(0.f, 0.f, 0.f, 0.f);
  float  al = 0.f;
  if (is_last) { bs = *(const float4*)(bias + f0); al = alpha[0]; }

  float4 colacc = make_float4(0.f, 0.f, 0.f, 0.f);
  const int rowStart = blockIdx.x * 128 + wave * 16;

  for (int i = 0; i < 16; ++i) {
    int r = rowStart + i;
    if (r >= N) break;
    size_t base = (size_t)r * OUT_FT + f0;
    float4 av  = *(const float4*)(agg + base);
    float4 kv  = *(const float4*)(k   + base);
    float4 k0v = *(const float4*)(k0  + base);

    float4 kn;                                   // kn = agg + 0.5*k
    kn.x = fmaf(0.5f, kv.x, av.x);
    kn.y = fmaf(0.5f, kv.y, av.y);
    kn.z = fmaf(0.5f, kv.z, av.z);
    kn.w = fmaf(0.5f, kv.w, av.w);

    float p = kn.x * ikap.x + kn.y * ikap.y + kn.z * ikap.z + kn.w * ikap.w;
    p += __shfl_xor(p, 1, 32);
    p += __shfl_xor(p, 2, 32);
    p += __shfl_xor(p, 4, 32);
    p += __shfl_xor(p, 8, 32);
    p += __shfl_xor(p, 16, 32);                  // D for this row
    float invD = 1.0f / p;

    float4 o;                                    // kn/(D*kap) + 0.1kn + 0.2k0
    o.x = kn.x * ikap.x * invD + 0.1f * kn.x + 0.2f * k0v.x;
    o.y = kn.y * ikap.y * invD + 0.1f * kn.y + 0.2f * k0v.y;
    o.z = kn.z * ikap.z * invD + 0.1f * kn.z + 0.2f * k0v.z;
    o.w = kn.w * ikap.w * invD + 0.1f * kn.w + 0.2f * k0v.w;

    if (is_last) {                               // + bias, PReLU
      o.x += bs.x; o.y += bs.y; o.z += bs.z; o.w += bs.w;
      o.x = o.x >= 0.f ? o.x : al * o.x;
      o.y = o.y >= 0.f ? o.y : al * o.y;
      o.z = o.z >= 0.f ? o.z : al * o.z;
      o.w = o.w >= 0.f ? o.w : al * o.w;
    } else {
      colacc.x += o.x; colacc.y += o.y; colacc.z += o.z; colacc.w += o.w;
    }
    *(float4*)(k + base) = o;
  }

  if (!is_last) {
    atomicAdd(&cs[f0 + 0], colacc.x);            // ds_add_f32
    atomicAdd(&cs[f0 + 1], colacc.y);
    atomicAdd(&cs[f0 + 2], colacc.z);
    atomicAdd(&cs[f0 + 3], colacc.w);
    __syncthreads();
    if (threadIdx.x < OUT_FT) atomicAdd(&kap_next[threadIdx.x], cs[threadIdx.x]);
  }
}

// ---------------------------------------------------------------------------
extern "C" void kernel_launch(void* const* d_in, const int* in_sizes, int n_in,
                              void* d_out, int out_size, void* d_ws, size_t ws_size,
                              hipStream_t stream) {
  const float* seq   = (const float*)d_in[0];
  const int*   erow  = (const int*)  d_in[1];
  const int*   ecol  = (const int*)  d_in[2];
  const float* ew    = (const float*)d_in[3];
  const float* Wfc   = (const float*)d_in[4];
  const float* Wfc2  = (const float*)d_in[5];
  const float* mm    = (const float*)d_in[6];
  const float* bias  = (const float*)d_in[7];
  const float* alpha = (const float*)d_in[8];

  const int N = in_sizes[0] / IN_FT;
  const int E = in_sizes[1];

  float* k    = (float*)d_out;                   // evolving state lives in d_out
  float* k0   = (float*)d_ws;                    // residual    [N,128]
  float* agg  = k0  + (size_t)N * OUT_FT;        // scatter dst [N,128]
  float* kap  = agg + (size_t)N * OUT_FT;        // kappa ping-pong [2][128]

  (void)hipMemsetAsync(kap, 0, 2 * OUT_FT * sizeof(float), stream);

  dim3 blk(256);
  gcn_front_gemm<<<(N + 15) / 16, blk, 0, stream>>>(seq, Wfc, Wfc2, mm,
                                                    k, k0, kap, N);

  const int scBlocks  = (int)(((long long)E * OUT_FT + 255) / 256);
  const int updBlocks = (N + 127) / 128;

  for (int t = 0; t < 3; ++t) {
    (void)hipMemsetAsync(agg, 0, (size_t)N * OUT_FT * sizeof(float), stream);
    gcn_scatter<<<scBlocks, blk, 0, stream>>>(k, erow, ecol, ew, agg, E);

    const float* kcur  = (t == 1) ? kap + OUT_FT : kap;   // t0:kap0 t1:kap1 t2:kap0
    float*       knext = (t == 0) ? kap + OUT_FT : kap;
    gcn_update<<<updBlocks, blk, 0, stream>>>(k, agg, k0, kcur, knext,
                                              bias, alpha, N, (t == 2));
    if (t == 0) (void)hipMemsetAsync(kap, 0, OUT_FT * sizeof(float), stream);
  }
}